// StraightThroughSubsetSampler_70875550319400
// MI455X (gfx1250) — compile-verified
//
#include <hip/hip_runtime.h>
#include <cstdint>

#define N_COLS  10000
#define NCHUNK  2500          // 2500 x 16B chunks per 40000-B row (rows 16B-aligned)
#define BLOCK   1024
#define NPT     10            // ceil(10000/1024)
#define KSEL    8
#define EPSV    1e-8f
#define NEG_INF (-__builtin_inff())
#define L2E     1.4426950408889634f   // log2(e)
#define C_GUM   0.5287663729448977f   // -log2(ln 2)

__device__ __forceinline__ float wave_max(float v) {
#pragma unroll
  for (int m = 16; m > 0; m >>= 1) v = fmaxf(v, __shfl_xor(v, m, 32));
  return v;
}
__device__ __forceinline__ float wave_sum(float v) {
#pragma unroll
  for (int m = 16; m > 0; m >>= 1) v += __shfl_xor(v, m, 32);
  return v;
}

// One-barrier block reductions: every wave redundantly folds the 32 LDS
// partials (xor-shuffle leaves the result in all lanes). Caller must
// alternate the scratch buffer between consecutive reductions.
__device__ __forceinline__ float block_max1(float v, float* red) {
  const int lane = threadIdx.x & 31, wid = threadIdx.x >> 5;
  v = wave_max(v);
  if (lane == 0) red[wid] = v;
  __syncthreads();
  return wave_max(red[lane]);
}
__device__ __forceinline__ float block_sum1(float v, float* red) {
  const int lane = threadIdx.x & 31, wid = threadIdx.x >> 5;
  v = wave_sum(v);
  if (lane == 0) red[wid] = v;
  __syncthreads();
  return wave_sum(red[lane]);
}
__device__ __forceinline__ void block_argmax1(float v, int i, float* rv, int* ri,
                                              float& outv, int& outi) {
  const int lane = threadIdx.x & 31, wid = threadIdx.x >> 5;
#pragma unroll
  for (int m = 16; m > 0; m >>= 1) {
    float nv = __shfl_xor(v, m, 32); int ni = __shfl_xor(i, m, 32);
    if (nv > v || (nv == v && ni < i)) { v = nv; i = ni; }
  }
  if (lane == 0) { rv[wid] = v; ri[wid] = i; }
  __syncthreads();
  v = rv[lane]; i = ri[lane];
#pragma unroll
  for (int m = 16; m > 0; m >>= 1) {
    float nv = __shfl_xor(v, m, 32); int ni = __shfl_xor(i, m, 32);
    if (nv > v || (nv == v && ni < i)) { v = nv; i = ni; }
  }
  outv = v; outi = i;
}

extern "C" __global__ __launch_bounds__(BLOCK)
void st_subset_sampler_kernel(const float* __restrict__ scores,
                              const float* __restrict__ u,
                              float* __restrict__ out) {
  extern __shared__ char smem[];
  float* sScore = (float*)(smem);            // 40000 B
  float* sU     = (float*)(smem + 40000);    // 40000 B
  float* redv   = (float*)(smem + 80000);    // 2 x 32 floats (double-buffered)
  int*   redi   = (int*)  (smem + 80256);    // 2 x 32 ints

  const int tid = threadIdx.x;
  const int row = blockIdx.x;
  const size_t rowOff = (size_t)row * N_COLS;

  // ---- CDNA5 async DMA of both input rows into LDS (ASYNCcnt-tracked) ----
  const unsigned ldsBase = (unsigned)(uintptr_t)smem;  // low 32b = LDS byte offset
  const uint64_t gScore  = (uint64_t)(const void*)(scores + rowOff);
  const uint64_t gU      = (uint64_t)(const void*)(u + rowOff);

  for (int c = tid; c < NCHUNK; c += BLOCK) {
    unsigned la = ldsBase + (unsigned)c * 16u;
    uint64_t ga = gScore + (uint64_t)c * 16u;
    asm volatile("global_load_async_to_lds_b128 %0, %1, off"
                 :: "v"(la), "v"(ga) : "memory");
    unsigned lb = ldsBase + 40000u + (unsigned)c * 16u;
    uint64_t gb = gU + (uint64_t)c * 16u;
    asm volatile("global_load_async_to_lds_b128 %0, %1, off"
                 :: "v"(lb), "v"(gb) : "memory");
  }
  asm volatile("s_wait_asynccnt 0" ::: "memory");
  __syncthreads();

  int buf = 0;  // reduction scratch parity

  // ---- init in base-2 domain: s2 = log2e*(scores + gumbel(u)) ----
  // s2 = scores*log2e + (-log2(ln2)) - log2(-log2(u))
  float s2[NPT], relaxed[NPT], e[NPT];
  float lm = NEG_INF;
#pragma unroll
  for (int i = 0; i < NPT; i++) {
    const int idx = tid + i * BLOCK;
    relaxed[i] = 0.0f;
    if (idx < N_COLS) {
      float gl = __builtin_amdgcn_logf(-__builtin_amdgcn_logf(sU[idx]));
      s2[i] = fmaf(sScore[idx], L2E, C_GUM - gl);
      lm = fmaxf(lm, s2[i]);
    } else {
      s2[i] = NEG_INF;
    }
  }

  // Single stability bound for all K softmaxes: s2 only decreases
  // (each update adds log2(mask) <= 0), so exp2(s2 - m2) <= 1 forever.
  const float m2 = block_max1(lm, redv + 32 * buf); buf ^= 1;

  // ---- K relaxed softmax iterations (tau = 1) ----
#pragma unroll 1
  for (int t = 0; t < KSEL; t++) {
    float ls = 0.0f;
#pragma unroll
    for (int i = 0; i < NPT; i++) {
      e[i] = __builtin_amdgcn_exp2f(s2[i] - m2);  // -inf -> 0 for padded slots
      ls += e[i];
    }
    const float Z = block_sum1(ls, redv + 32 * buf); buf ^= 1;
    const float inv = __builtin_amdgcn_rcpf(Z);
#pragma unroll
    for (int i = 0; i < NPT; i++) {
      float ohv = e[i] * inv;
      relaxed[i] += ohv;
      if (t + 1 < KSEL)
        s2[i] += __builtin_amdgcn_logf(fmaxf(1.0f - ohv, EPSV));
    }
  }

  // ---- hard top-8: incremental per-thread best + 1-barrier block argmax ----
  unsigned khot = 0u;  // bit i: my slot i is selected
  float bv; int bi;
  auto rescan = [&]() {
    bv = NEG_INF; bi = 0x7fffffff;
#pragma unroll
    for (int i = 0; i < NPT; i++) {
      const int idx = tid + i * BLOCK;
      const bool ok = (idx < N_COLS) && !((khot >> i) & 1u);
      const float v = ok ? relaxed[i] : NEG_INF;
      if (v > bv) { bv = v; bi = idx; }  // ascending idx => first occurrence kept
    }
  };
  rescan();

#pragma unroll 1
  for (int t = 0; t < KSEL; t++) {
    float wv; int wi;
    block_argmax1(bv, bi, redv + 32 * buf, redi + 32 * buf, wv, wi); buf ^= 1;
    if ((wi & (BLOCK - 1)) == tid) {     // I own the winner
      khot |= 1u << (wi >> 10);
      if (t + 1 < KSEL) rescan();        // only the owner rescans its 10 slots
    }
  }

  // ---- straight-through output: (hard - sample) + sample ----
#pragma unroll
  for (int i = 0; i < NPT; i++) {
    const int idx = tid + i * BLOCK;
    if (idx < N_COLS) {
      const float hard = ((khot >> i) & 1u) ? 1.0f : 0.0f;
      out[rowOff + idx] = (hard - relaxed[i]) + relaxed[i];
    }
  }
}

extern "C" void kernel_launch(void* const* d_in, const int* in_sizes, int n_in,
                              void* d_out, int out_size, void* d_ws, size_t ws_size,
                              hipStream_t stream) {
  const float* scores = (const float*)d_in[0];
  const float* u      = (const float*)d_in[1];
  float* out = (float*)d_out;

  const int rows = in_sizes[0] / N_COLS;       // 4096
  const size_t shmem = 80000u + 512u;          // row buffers + double-buffered scratch

  st_subset_sampler_kernel<<<dim3(rows), dim3(BLOCK), shmem, stream>>>(scores, u, out);
}